// DemodulatedLinear_52003464020413
// MI455X (gfx1250) — compile-verified
//
#include <hip/hip_runtime.h>
#include <math.h>

typedef __attribute__((ext_vector_type(2))) float v2f;
typedef __attribute__((ext_vector_type(8))) float v8f;

#define IN_DIM    512
#define OUT_DIM   512
#define MOD_DIM   256
#define BATCH     1024
#define EPS       1e-8f
#define XS_STRIDE 516   // 516 mod 64 = 4 -> adjacent rows 4 banks apart (conflict-free frag reads)

// colsq[i] = sum_o weight[o,i]^2   (coalesced: thread i walks column i)
__global__ void dml_colsq_kernel(const float* __restrict__ weight,
                                 float* __restrict__ colsq) {
    int i = blockIdx.x * blockDim.x + threadIdx.x;
    if (i >= IN_DIM) return;
    float s = 0.f;
#pragma unroll 8
    for (int o = 0; o < OUT_DIM; ++o) {
        float w = weight[o * IN_DIM + i];
        s = fmaf(w, w, s);
    }
    colsq[i] = s;
}

// Fused: GEMM1 (scales) + demodulation -> LDS xs[16][512] -> GEMM2 (out).
// One block per 16-row batch tile; 8 waves; each wave owns 4 N-tiles per phase
// with a shared A fragment (4 wmma per A-load).
__global__ void __launch_bounds__(256)
dml_fused_kernel(const float* __restrict__ mod,    // [1024,256]
                 const float* __restrict__ mod_w,  // [512,256]
                 const float* __restrict__ mod_b,  // [512]
                 const float* __restrict__ x,      // [1024,512]
                 const float* __restrict__ colsq,  // [512]
                 const float* __restrict__ weight, // [512,512]
                 const float* __restrict__ bias,   // [512]
                 float* __restrict__ out) {        // [1024,512]
    __shared__ float lds_xs[16 * XS_STRIDE];       // ~32.3 KB of the 320 KB WGP LDS

    const int lane = threadIdx.x & 31;
    const int wave = threadIdx.x >> 5;             // 0..7
    const int tM   = blockIdx.x;                   // batch tile 0..63
    const int row  = lane & 15;                    // A-row / B-row / C-col index
    const int koff = (lane >> 4) << 1;             // 0 or 2 (K pair within step of 4)
    const int mloc = (lane >> 4) << 3;             // 0 or 8 (C fragment local row base)

    // ---------------- Phase 1: scales GEMM + demod -> LDS ----------------
    {
        const float* aRow = mod + (size_t)(tM * 16 + row) * MOD_DIM + koff;
        const float* bRow[4];
        v8f acc[4] = {};
#pragma unroll
        for (int t = 0; t < 4; ++t)
            bRow[t] = mod_w + (size_t)((wave * 4 + t) * 16 + row) * MOD_DIM + koff;

#pragma unroll 4
        for (int k = 0; k < MOD_DIM; k += 4) {
            v2f a = *(const v2f*)(aRow + k);
#pragma unroll
            for (int t = 0; t < 4; ++t) {
                v2f b = *(const v2f*)(bRow[t] + k);
                acc[t] = __builtin_amdgcn_wmma_f32_16x16x4_f32(
                    false, a, false, b, (short)0, acc[t], false, false);
            }
        }

        // s = acc + mod_b[i]; xs = x * s * rsqrt(s^2*colsq[i] + eps) -> LDS
#pragma unroll
        for (int t = 0; t < 4; ++t) {
            const int   gi = (wave * 4 + t) * 16 + row;  // global i (per-lane constant)
            const float mb = mod_b[gi];
            const float cs = colsq[gi];
#pragma unroll
            for (int r = 0; r < 8; ++r) {
                const int   mm = mloc + r;               // local batch row 0..15
                const float s  = acc[t][r] + mb;
                const float d  = rsqrtf(fmaf(s * s, cs, EPS));
                lds_xs[mm * XS_STRIDE + gi] =
                    x[(size_t)(tM * 16 + mm) * IN_DIM + gi] * s * d;
            }
        }
    }

    __syncthreads();

    // ---------------- Phase 2: out = xs @ weight^T + bias ----------------
    {
        const float* aLds = &lds_xs[row * XS_STRIDE + koff];  // ds_load_b64 fragments
        const float* bRow[4];
        v8f acc[4];
#pragma unroll
        for (int t = 0; t < 4; ++t) {
            const int   gn = (wave * 4 + t) * 16 + row;
            const float bv = bias[gn];
            acc[t] = (v8f){bv, bv, bv, bv, bv, bv, bv, bv};
            bRow[t] = weight + (size_t)gn * IN_DIM + koff;
        }

#pragma unroll 4
        for (int k = 0; k < IN_DIM; k += 4) {
            v2f a = *(const v2f*)(aLds + k);
#pragma unroll
            for (int t = 0; t < 4; ++t) {
                v2f b = *(const v2f*)(bRow[t] + k);
                acc[t] = __builtin_amdgcn_wmma_f32_16x16x4_f32(
                    false, a, false, b, (short)0, acc[t], false, false);
            }
        }

#pragma unroll
        for (int t = 0; t < 4; ++t) {
            const int gn = (wave * 4 + t) * 16 + row;
#pragma unroll
            for (int r = 0; r < 8; ++r)
                out[(size_t)(tM * 16 + mloc + r) * OUT_DIM + gn] = acc[t][r];
        }
    }
}

extern "C" void kernel_launch(void* const* d_in, const int* in_sizes, int n_in,
                              void* d_out, int out_size, void* d_ws, size_t ws_size,
                              hipStream_t stream) {
    const float* mod    = (const float*)d_in[0]; // [1024,256]
    const float* x      = (const float*)d_in[1]; // [1024,512]
    const float* weight = (const float*)d_in[2]; // [512,512]
    const float* bias   = (const float*)d_in[3]; // [512]
    const float* mod_w  = (const float*)d_in[4]; // [512,256]
    const float* mod_b  = (const float*)d_in[5]; // [512]
    float* out = (float*)d_out;                  // [1024,512]

    float* colsq = (float*)d_ws;                 // 512 floats of scratch

    dml_colsq_kernel<<<(IN_DIM + 255) / 256, 256, 0, stream>>>(weight, colsq);

    dml_fused_kernel<<<BATCH / 16, 256, 0, stream>>>(
        mod, mod_w, mod_b, x, colsq, weight, bias, out);
}